// InnerProductNetwork_86517821214535
// MI455X (gfx1250) — compile-verified
//
#include <hip/hip_runtime.h>

// InnerProductNetwork pairwise interactions as batched Gram matrices via
// V_WMMA_F32_16X16X4_F32 (CDNA5 / gfx1250, wave32).
//
// Per batch element b: G = X @ X^T, X = (32 fields x 16 dims, f32);
// out[b, p] = G[i, j] for i<j in row-major pair order (496 pairs).
//
// One wave per batch element:
//   X0 = rows 0..15, X1 = rows 16..31
//   G00 = X0*X0^T (strict upper), G01 = X0*X1^T (all), G11 = X1*X1^T (strict upper)
//   each tile = 4 chained wmma_f32_16x16x4_f32 (K = 16 in 4 steps of 4)
//
// Memory-bound: 33.5 MB in + 32.5 MB out @ 23.3 TB/s ~= 2.8 us floor; the
// instruction stream is kept ~2x under that. b is wave-uniform via
// readfirstlane so all global accesses use SADDR (SGPR base + 32-bit VGPR
// offset) addressing; pair indices are add-only recurrences.

typedef __attribute__((ext_vector_type(2))) float v2f;
typedef __attribute__((ext_vector_type(8))) float v8f;

__global__ __launch_bounds__(256) void ipn_gram_wmma(const float* __restrict__ x,
                                                     float* __restrict__ out,
                                                     int batch) {
  const int lane = threadIdx.x & 31;   // wave32
  const int wave = threadIdx.x >> 5;   // 8 waves / block
  // wave-uniform batch index, forced into an SGPR
  const int b = __builtin_amdgcn_readfirstlane(blockIdx.x * 8 + wave);
  if (b >= batch) return;              // scalar branch, EXEC untouched

  const int half = lane >> 4;          // K-phase half (ISA A/B striping)
  const int r    = lane & 15;          // row of X0/X1 this lane carries (M for A, N for B)

  const float* __restrict__ xb = x + (size_t)b * 512;  // SGPR base
  const int off = r * 16 + 2 * half;                   // 32-bit lane offset (elements)

  // A 16x4 f32 layout: lanes 0-15 hold K=4k+0,4k+1; lanes 16-31 hold K=4k+2,4k+3.
  // B 4x16 mirrors with lane = N. Both operands are per-lane rows of X.
  v2f a0[4], a1[4];
#pragma unroll
  for (int k = 0; k < 4; ++k) {
    a0[k] = *(const v2f*)(xb + off + 4 * k);        // X0 row r   (global_load_b64, saddr)
    a1[k] = *(const v2f*)(xb + off + 256 + 4 * k);  // X1 row r
  }

  v8f c00 = {}, c01 = {}, c11 = {};
#pragma unroll
  for (int k = 0; k < 4; ++k) {
    // (neg_a, A, neg_b, B, c_mod, C, reuse_a, reuse_b)
    c00 = __builtin_amdgcn_wmma_f32_16x16x4_f32(false, a0[k], false, a0[k],
                                                (short)0, c00, false, false);
    c01 = __builtin_amdgcn_wmma_f32_16x16x4_f32(false, a0[k], false, a1[k],
                                                (short)0, c01, false, false);
    c11 = __builtin_amdgcn_wmma_f32_16x16x4_f32(false, a1[k], false, a1[k],
                                                (short)0, c11, false, false);
  }

  // C/D layout: VGPR v, lane L -> M = v + 8*(L>=16), N = L%16.
  float* __restrict__ ob = out + (size_t)b * 496;   // SGPR base
  const int mbase = half * 8;
  // Pair index p(i,j) = 31i - i(i-1)/2 + (j-i-1); start(0)=0, start(8)=220,
  // start(16)=376, start(24)=468. Add-only recurrences (verified):
  //   p01(v): (i=mbase+v,   j=r+16) init (half?212:0)+r+15,  step += 30-m
  //   p00(v): (i=mbase+v,   j=r   ) = p01(v) - 16 (folds into store offset)
  //   p11(v): (i=mbase+16+v,j=r+16) init (half?460:376)+r-1, step += 14-m
  int p01 = (half ? 212 : 0) + r + 15;
  int p11 = (half ? 460 : 376) + r - 1;
#pragma unroll
  for (int v = 0; v < 8; ++v) {
    const int m = mbase + v;
    ob[p01] = c01[v];              // always i<j
    if (r > m) {                   // strict upper of both diagonal tiles
      ob[p01 - 16] = c00[v];       // p00 = p01 - 16
      ob[p11]      = c11[v];
    }
    p01 += 30 - m;
    p11 += 14 - m;
  }
}

extern "C" void kernel_launch(void* const* d_in, const int* in_sizes, int n_in,
                              void* d_out, int out_size, void* d_ws, size_t ws_size,
                              hipStream_t stream) {
  const float* x = (const float*)d_in[0];
  float* out = (float*)d_out;
  const int batch = in_sizes[0] / (32 * 16);          // 16384
  const int waves_per_block = 8;                      // 256 threads, wave32
  const int blocks = (batch + waves_per_block - 1) / waves_per_block;
  ipn_gram_wmma<<<blocks, 256, 0, stream>>>(x, out, batch);
}